// QuadraticScalarProductAttentionModule_58548994179562
// MI455X (gfx1250) — compile-verified
//
#include <hip/hip_runtime.h>
#include <stdint.h>

// B=16, S=2048, D=64, OUT=128
//   ref: out[b,s,o] = sum_t (X[b,s,:]·X[b,t,:]) W[o,t] + bias[o]
// Factored:
//   Stage 1: Y[b] = W @ X[b]            -> [128, 64], stored transposed Yt[b][d][o]
//   Stage 2: out[b] = X[b] @ Yt[b] + b  -> [2048, 128]
// Memory-bound (~26 MB HBM floor @ 23.3 TB/s); f32 WMMA (16x16x4) preserves
// reference fp32 numerics. B operands staged in LDS via async-to-LDS copies.

#define B_N   16
#define S_LEN 2048
#define D_DIM 64
#define OUT_F 128
#define KCH   64                // stage-1 K chunk (rows of X per LDS buffer)
#define S1_STRIDE (D_DIM + 8)   // 72: 2*72 mod 64 = 16 -> half-waves hit disjoint banks
#define S2_STRIDE (OUT_F + 8)   // 136: 2*136 mod 64 = 16

typedef __attribute__((ext_vector_type(2))) float v2f;
typedef __attribute__((ext_vector_type(8))) float v8f;

#if defined(__gfx1250__) && __has_builtin(__builtin_amdgcn_global_load_async_to_lds_b128)
#define USE_ASYNC 1
#else
#define USE_ASYNC 0
#endif

#if USE_ASYNC
// Exact pointee type from the clang diagnostic: GCC-vector int4.
typedef int vs4i __attribute__((vector_size(4 * sizeof(int))));
typedef __attribute__((address_space(1))) vs4i* gptr4i;   // global (AS1)
typedef __attribute__((address_space(3))) vs4i* lptr4i;   // LDS (AS3, 32-bit)
#endif

// Copy 16 bytes global -> LDS. Async (ASYNCcnt) when available, else sync fallback.
__device__ __forceinline__ void copy16B(float* lds_dst, const float* gsrc) {
#if USE_ASYNC
  __builtin_amdgcn_global_load_async_to_lds_b128(
      (gptr4i)(uintptr_t)gsrc,
      (lptr4i)(unsigned)(uintptr_t)lds_dst,   // flat addr[31:0] == LDS byte offset
      /*offset=*/0, /*cpol=*/0);
#else
  *(float4*)lds_dst = *(const float4*)gsrc;
#endif
}

__device__ __forceinline__ void async_wait0() {
#if USE_ASYNC
#if __has_builtin(__builtin_amdgcn_s_wait_asynccnt)
  __builtin_amdgcn_s_wait_asynccnt(0);
#else
  asm volatile("s_wait_asynccnt 0x0" ::: "memory");
#endif
#endif
}

// ---------------------------------------------------------------------------
// Stage 1: Y[b] = W @ X[b].  One block per batch; 8 waves = 8 o-tiles.
// X[b] streamed through double-buffered LDS (64x64 chunk, padded stride 72),
// copy of chunk c+1 overlapped with WMMAs on chunk c. A operand (W rows) via
// per-lane b64 global loads + prefetch. D stored transposed -> Yt[b][d][o].
// ---------------------------------------------------------------------------
__global__ __launch_bounds__(256) void qspa_stage1_wxt(
    const float* __restrict__ X, const float* __restrict__ W,
    float* __restrict__ Yt) {
  __shared__ __align__(16) float sX[2][KCH * S1_STRIDE];   // 2 x 18 KB

  const int batch = blockIdx.x;
  const int tid   = threadIdx.x;
  const int otile = tid >> 5;          // wave id = o-tile, 0..7
  const int lane  = tid & 31;
  const int half  = lane >> 4;         // 0: K=0,1   1: K=2,3
  const int l16   = lane & 15;

  const float* __restrict__ Xb   = X + batch * (S_LEN * D_DIM);
  const float* __restrict__ Wrow = W + (otile * 16 + l16) * S_LEN;

  // cooperative chunk copy: thread -> (row cr, 16-float slice cq); 4 x b128 each
  const int cr = tid >> 2;             // 0..63
  const int cq = (tid & 3) * 16;       // 0,16,32,48

  v8f acc[4] = {};

  // kick chunk 0
  {
    const float* src = Xb + cr * D_DIM + cq;
    float* dst = &sX[0][cr * S1_STRIDE + cq];
#pragma unroll
    for (int j = 0; j < 4; ++j) copy16B(dst + 4 * j, src + 4 * j);
  }
  async_wait0();
  __syncthreads();

  const int NCH = S_LEN / KCH;         // 32 chunks
  for (int c = 0; c < NCH; ++c) {
    const int cur = c & 1;
    if (c + 1 < NCH) {                 // issue async copy of next chunk
      const float* src = Xb + (c + 1) * (KCH * D_DIM) + cr * D_DIM + cq;
      float* dst = &sX[cur ^ 1][cr * S1_STRIDE + cq];
#pragma unroll
      for (int j = 0; j < 4; ++j) copy16B(dst + 4 * j, src + 4 * j);
    }

    const float* wbase = Wrow + c * KCH;
    __builtin_prefetch(wbase + KCH, 0, 1);   // global_prefetch_b8 on W stream
    const float* sbuf = &sX[cur][0];

#pragma unroll 4
    for (int kk = 0; kk < KCH; kk += 4) {
      // A frag 16x4: lanes 0-15 -> A[M=l16][kk..kk+1], lanes 16-31 -> kk+2..kk+3
      v2f a = *(const v2f*)(wbase + kk + 2 * half);
#pragma unroll
      for (int nt = 0; nt < 4; ++nt) {
        // B frag 4x16 from LDS: v0 rows kk / kk+2, v1 rows kk+1 / kk+3
        const float* bp = sbuf + (kk + 2 * half) * S1_STRIDE + nt * 16 + l16;
        v2f bb;
        bb.x = bp[0];
        bb.y = bp[S1_STRIDE];
        acc[nt] = __builtin_amdgcn_wmma_f32_16x16x4_f32(
            false, a, false, bb, (short)0, acc[nt], false, false);
      }
    }

    async_wait0();     // next chunk landed (overlapped with compute above)
    __syncthreads();   // all waves done with buf[cur]; copies visible to all
  }

  // D layout: VGPR v, lanes 0-15 -> (M=v, N=l16); lanes 16-31 -> (M=v+8, N=l16).
  // Store transposed Yt[d=N][o=M]: 8 o-contiguous floats per lane (2x b128).
  float* __restrict__ Ybase = Yt + batch * (D_DIM * OUT_F);
#pragma unroll
  for (int nt = 0; nt < 4; ++nt) {
    const int d = nt * 16 + l16;
    const int o = otile * 16 + half * 8;
    *(v8f*)(Ybase + d * OUT_F + o) = acc[nt];
  }
}

// ---------------------------------------------------------------------------
// Stage 2: out[b] = X[b] @ Yt[b] + bias.  One block per (batch, 8 s-tiles);
// Yt[b] (32 KB) async-copied to LDS once (padded stride 136), B frags via ds.
// ---------------------------------------------------------------------------
__global__ __launch_bounds__(256) void qspa_stage2_out(
    const float* __restrict__ X, const float* __restrict__ Yt,
    const float* __restrict__ bias, float* __restrict__ out) {
  __shared__ __align__(16) float sY[D_DIM * S2_STRIDE];    // 34 KB

  const int batch = blockIdx.x >> 4;     // 16 blocks per batch
  const int grp   = blockIdx.x & 15;
  const int tid   = threadIdx.x;
  const int wid   = tid >> 5;
  const int stile = grp * 8 + wid;       // 0..127
  const int lane  = tid & 31;
  const int half  = lane >> 4;
  const int l16   = lane & 15;

  // cooperative Yt copy: thread -> (row cr, 32-float slice cq); 8 x b128 each
  {
    const int cr = tid >> 2;             // 0..63
    const int cq = (tid & 3) * 32;       // 0,32,64,96
    const float* src = Yt + batch * (D_DIM * OUT_F) + cr * OUT_F + cq;
    float* dst = &sY[cr * S2_STRIDE + cq];
#pragma unroll
    for (int j = 0; j < 8; ++j) copy16B(dst + 4 * j, src + 4 * j);
  }
  async_wait0();
  __syncthreads();

  const float* __restrict__ Xrow =
      X + batch * (S_LEN * D_DIM) + (stile * 16 + l16) * D_DIM;

  v8f acc[8] = {};

#pragma unroll 4
  for (int k = 0; k < D_DIM; k += 4) {
    v2f a = *(const v2f*)(Xrow + k + 2 * half);
#pragma unroll
    for (int nt = 0; nt < 8; ++nt) {
      const float* bp = &sY[(k + 2 * half) * S2_STRIDE + nt * 16 + l16];
      v2f bb;
      bb.x = bp[0];
      bb.y = bp[S2_STRIDE];
      acc[nt] = __builtin_amdgcn_wmma_f32_16x16x4_f32(
          false, a, false, bb, (short)0, acc[nt], false, false);
    }
  }

  // bias (per o-column, constant across v) + store
  float* __restrict__ Obase = out + batch * (S_LEN * OUT_F) + (stile * 16) * OUT_F;
#pragma unroll
  for (int nt = 0; nt < 8; ++nt) {
    const float bv = bias[nt * 16 + l16];
#pragma unroll
    for (int v = 0; v < 8; ++v) {
      const int m = v + half * 8;
      Obase[m * OUT_F + nt * 16 + l16] = acc[nt][v] + bv;
    }
  }
}

// ---------------------------------------------------------------------------
extern "C" void kernel_launch(void* const* d_in, const int* in_sizes, int n_in,
                              void* d_out, int out_size, void* d_ws, size_t ws_size,
                              hipStream_t stream) {
  const float* X    = (const float*)d_in[0];   // [16, 2048, 64]
  const float* W    = (const float*)d_in[1];   // [128, 2048]
  const float* bias = (const float*)d_in[2];   // [128]
  float* out = (float*)d_out;                  // [16, 2048, 128]
  float* Yt  = (float*)d_ws;                   // [16, 64, 128] = 512 KB scratch

  qspa_stage1_wxt<<<dim3(B_N), dim3(256), 0, stream>>>(X, W, Yt);
  qspa_stage2_out<<<dim3(B_N * 16), dim3(256), 0, stream>>>(X, Yt, bias, out);
}